// SVP_88862873354276
// MI455X (gfx1250) — compile-verified
//
#include <hip/hip_runtime.h>
#include <math.h>

// out = softmax(x @ W^T + b), x:[1024,4096] f32, W:[50257,4096] f32, b:[50257]
// Memory-bound: W stream (823MB) dominates -> fp32 WMMA (V_WMMA_F32_16X16X4_F32),
// double-buffered GLOBAL_LOAD_ASYNC_TO_LDS_B128 staging (ASYNCcnt pipeline).

#define B_ROWS 1024
#define D_DIM  4096
#define C_DIM  50257

#define BM 64
#define BN 64
#define BK 64
#define LDSS 68   // padded row stride (floats): 16B-aligned, conflict-free lane-strided reads
#define NCHUNK (D_DIM / BK)

typedef __attribute__((ext_vector_type(2))) float v2f;
typedef __attribute__((ext_vector_type(8))) float v8f;

// Async copy 16B global -> LDS (per-lane), tracked by ASYNCcnt.
__device__ __forceinline__ void async_ld16(unsigned lds_off, unsigned goff, const void* base)
{
    asm volatile("global_load_async_to_lds_b128 %0, %1, %2"
                 :: "v"(lds_off), "v"(goff), "s"(base)
                 : "memory");
}

// Issue one K-chunk (64x64 x-tile + 64x64 W-tile) as 8 async B128 loads per thread.
__device__ __forceinline__ void stage_chunk_async(const float* __restrict__ x,
                                                  const float* __restrict__ W,
                                                  unsigned ldsx, unsigned ldsw,
                                                  int m0, int n0, int k0, int tid)
{
#pragma unroll
    for (int i = 0; i < 4; ++i) {
        const int q   = tid + i * 256;   // 0..1023
        const int row = q >> 4;          // 0..63
        const int c4  = (q & 15) << 2;   // 0,4,..,60
        const unsigned loff = (unsigned)((row * LDSS + c4) * 4);

        const unsigned gx = (unsigned)(((m0 + row) * D_DIM + k0 + c4) * 4);
        async_ld16(ldsx + loff, gx, x);

        // clamp OOB class rows: garbage corrupts only columns >= C, masked at store
        int wrow = n0 + row;
        if (wrow >= C_DIM) wrow = C_DIM - 1;
        const unsigned gw = (unsigned)(((size_t)wrow * D_DIM + (size_t)(k0 + c4)) * 4);
        async_ld16(ldsw + loff, gw, W);
    }
}

__global__ __launch_bounds__(256)
void svp_gemm_wmma_f32(const float* __restrict__ x,
                       const float* __restrict__ W,
                       const float* __restrict__ bias,
                       float* __restrict__ out)
{
    __shared__ __align__(16) float lds_x[2][BM * LDSS];
    __shared__ __align__(16) float lds_w[2][BN * LDSS];

    const int tid  = threadIdx.x;
    const int wave = tid >> 5;      // 8 waves of 32 (wave32)
    const int lane = tid & 31;
    const int half = lane >> 4;     // 0: lanes 0-15, 1: lanes 16-31
    const int lm   = lane & 15;

    const int m0 = blockIdx.x * BM; // M tile (fast grid dim -> same-N blocks adjacent, W hits L2)
    const int n0 = blockIdx.y * BN; // N tile (classes)

    // wave sub-tile: 16 rows (M) x 32 cols (N) => 2 accumulators
    const int m_sub = (wave & 3) * 16;
    const int n_sub = (wave >> 2) * 32;

    const unsigned ldsx[2] = { (unsigned)(size_t)&lds_x[0][0], (unsigned)(size_t)&lds_x[1][0] };
    const unsigned ldsw[2] = { (unsigned)(size_t)&lds_w[0][0], (unsigned)(size_t)&lds_w[1][0] };

    v8f acc0 = {0.f, 0.f, 0.f, 0.f, 0.f, 0.f, 0.f, 0.f};
    v8f acc1 = {0.f, 0.f, 0.f, 0.f, 0.f, 0.f, 0.f, 0.f};

    // prologue: stage chunk 0 into buffer 0
    stage_chunk_async(x, W, ldsx[0], ldsw[0], m0, n0, 0, tid);

    for (int i = 0; i < NCHUNK; ++i) {
        const int cur = i & 1;
        // issue next chunk into the other buffer (last iter re-stages chunk 0: harmless, unread)
        const int k_next = ((i + 1) % NCHUNK) * BK;
        stage_chunk_async(x, W, ldsx[1 - cur], ldsw[1 - cur], m0, n0, k_next, tid);

        // drain the PREVIOUS 8 async ops (in-order per wave) => current buffer complete
        asm volatile("s_wait_asynccnt 8" ::: "memory");
        __syncthreads();   // all waves' chunk-i LDS writes visible

        // A 16x4 f32 layout: lanes0-15 VGPR{0,1}=K{0,1}; lanes16-31=K{2,3} -> koff=2*half
        // B 4x16 f32 layout: mirrored, N = lane%16
        const float* ap  = &lds_x[cur][(m_sub + lm) * LDSS];
        const float* b0p = &lds_w[cur][(n_sub + lm) * LDSS];
        const float* b1p = &lds_w[cur][(n_sub + 16 + lm) * LDSS];
        const int koff = 2 * half;
#pragma unroll
        for (int kk = 0; kk < BK; kk += 4) {
            const v2f a  = *(const v2f*)(ap  + kk + koff);  // 8B-aligned (even float idx)
            const v2f b0 = *(const v2f*)(b0p + kk + koff);
            const v2f b1 = *(const v2f*)(b1p + kk + koff);
            acc0 = __builtin_amdgcn_wmma_f32_16x16x4_f32(
                       false, a, false, b0, (short)0, acc0, false, false);
            acc1 = __builtin_amdgcn_wmma_f32_16x16x4_f32(
                       false, a, false, b1, (short)0, acc1, false, false);
        }
        __syncthreads();   // all waves done reading buf[cur] before it is overwritten
    }

    // C/D layout: VGPR r, lanes0-15 -> (M=r, N=lane); lanes16-31 -> (M=r+8, N=lane-16)
    // => per-VGPR store is two contiguous 64B segments (coalesced).
    const int gm_base = m0 + m_sub + 8 * half;
    const int gn0 = n0 + n_sub + lm;
    const int gn1 = gn0 + 16;
    const float bi0 = (gn0 < C_DIM) ? bias[gn0] : 0.f;
    const float bi1 = (gn1 < C_DIM) ? bias[gn1] : 0.f;
#pragma unroll
    for (int r = 0; r < 8; ++r) {
        const int gm = gm_base + r;
        if (gn0 < C_DIM) out[(size_t)gm * C_DIM + gn0] = acc0[r] + bi0;
        if (gn1 < C_DIM) out[(size_t)gm * C_DIM + gn1] = acc1[r] + bi1;
    }
}

// In-place row softmax over C=50257: one block per row.
// Pass 1: single-read online (max,sum); LDS tree reduce; Pass 2: normalize (read+write).
__global__ __launch_bounds__(256)
void svp_softmax_rows(float* __restrict__ out)
{
    const int tid = threadIdx.x;
    float* row = out + (size_t)blockIdx.x * C_DIM;

    __shared__ float sm_m[256];
    __shared__ float sm_s[256];

    float mx = -INFINITY;
    float sm = 0.f;
    for (int c = tid; c < C_DIM; c += 256) {
        const float v  = row[c];
        const float nm = fmaxf(mx, v);
        sm = sm * __expf(mx - nm) + __expf(v - nm);
        mx = nm;
    }
    sm_m[tid] = mx;
    sm_s[tid] = sm;
    __syncthreads();

    for (int st = 128; st > 0; st >>= 1) {
        if (tid < st) {
            const float m1 = sm_m[tid],      s1 = sm_s[tid];
            const float m2 = sm_m[tid + st], s2 = sm_s[tid + st];
            const float M  = fmaxf(m1, m2);
            sm_s[tid] = s1 * __expf(m1 - M) + s2 * __expf(m2 - M);
            sm_m[tid] = M;
        }
        __syncthreads();
    }

    const float M   = sm_m[0];
    const float inv = 1.f / sm_s[0];
    for (int c = tid; c < C_DIM; c += 256)
        row[c] = __expf(row[c] - M) * inv;
}

extern "C" void kernel_launch(void* const* d_in, const int* in_sizes, int n_in,
                              void* d_out, int out_size, void* d_ws, size_t ws_size,
                              hipStream_t stream) {
    (void)in_sizes; (void)n_in; (void)d_ws; (void)ws_size; (void)out_size;
    const float* x  = (const float*)d_in[0];
    const float* W  = (const float*)d_in[1];
    const float* b  = (const float*)d_in[2];
    float* out = (float*)d_out;

    dim3 grid(B_ROWS / BM, (C_DIM + BN - 1) / BN);   // (16, 786), M fastest
    svp_gemm_wmma_f32<<<grid, 256, 0, stream>>>(x, W, b, out);
    svp_softmax_rows<<<B_ROWS, 256, 0, stream>>>(out);
}